// RNNCellModel_14929306321663
// MI455X (gfx1250) — compile-verified
//
#include <hip/hip_runtime.h>

typedef __attribute__((ext_vector_type(2))) float v2f;
typedef __attribute__((ext_vector_type(8))) float v8f;

#define S_LEN 512
#define BATCH 32768
#define ISZ   3
#define HSZ   5

__device__ __forceinline__ float fast_tanh(float v) {
#if __has_builtin(__builtin_amdgcn_tanhf)
    return __builtin_amdgcn_tanhf(v);      // v_tanh_f32 (single TRANS op)
#else
    float e = __expf(2.0f * v);            // v_exp_f32 path
    return 1.0f - 2.0f / (e + 1.0f);
#endif
}

// One wave (32 lanes) owns 16 batch rows; hidden state lives in registers in
// WMMA D-layout (m across v8f elements 0..4, batch col n = lane 0..15).
// Each step: D = A1*B1 + (A2*B2 + C) chained as two v_wmma_f32_16x16x4_f32,
// where A1 = W_hh[:,0:4], A2 = {W_hh[:,4], W_ih[:,0..2]}, C = bias rows.
__global__ void __launch_bounds__(32)
rnn_scan_wmma(const float* __restrict__ x,
              const float* __restrict__ hx0,
              const float* __restrict__ Wih,
              const float* __restrict__ Whh,
              const float* __restrict__ bih,
              const float* __restrict__ bhh,
              float* __restrict__ out)
{
    const int  lane = threadIdx.x;        // 0..31
    const int  col  = lane & 15;          // batch column within tile / row m
    const bool lo   = lane < 16;
    const int  n    = blockIdx.x * 16 + col;

    // ---- A-matrix operands (16x4 f32 layout: lanes0-15 hold K0,K1; lanes16-31 K2,K3)
    const int m = col;
    float a1v0 = 0.f, a1v1 = 0.f, a2v0 = 0.f, a2v1 = 0.f;
    if (m < HSZ) {
        a1v0 = Whh[m * HSZ + (lo ? 0 : 2)];          // K0 / K2 of W_hh[:,0:4]
        a1v1 = Whh[m * HSZ + (lo ? 1 : 3)];          // K1 / K3
        a2v0 = lo ? Whh[m * HSZ + 4] : Wih[m * ISZ + 1]; // K0 = Whh col4, K2 = Wih col1
        a2v1 = lo ? Wih[m * ISZ + 0] : Wih[m * ISZ + 2]; // K1 = Wih col0, K3 = Wih col2
    }
    const v2f A1 = {a1v0, a1v1};
    const v2f A2 = {a2v0, a2v1};

    // ---- bias in C-layout: element r, lanes0-15 = (b_ih+b_hh)[r]; everything else 0
    v8f cbias = {};
    #pragma unroll
    for (int r = 0; r < HSZ; ++r)
        cbias[r] = lo ? (bih[r] + bhh[r]) : 0.f;

    // ---- initial hidden state (valid in lanes 0..15)
    float hr0 = hx0[n * HSZ + 0];
    float hr1 = hx0[n * HSZ + 1];
    float hr2 = hx0[n * HSZ + 2];
    float hr3 = hx0[n * HSZ + 3];
    float hr4 = hx0[n * HSZ + 4];

    const int xoffA = lo ? 0 : 1;  // upper lanes fetch x1 (B2 K2 row)
    const int xoffB = lo ? 0 : 2;  // lower: x0 (B2 K1 row), upper: x2 (B2 K3 row)

    for (int t = 0; t < S_LEN; ++t) {
        const float* xp = x + ((size_t)t * BATCH + n) * ISZ;
        float xA = __builtin_nontemporal_load(xp + xoffA);
        float xB = __builtin_nontemporal_load(xp + xoffB);
        if (t + 8 < S_LEN)   // stream-ahead prefetch (global_prefetch_b8)
            __builtin_prefetch(x + ((size_t)(t + 8) * BATCH + n) * ISZ, 0, 1);

        // rebuild B-layout operands (4x16: lanes0-15 = K0,K1; lanes16-31 = K2,K3)
        float s2 = __shfl(hr2, col, 32);            // h2 -> upper-lane K2 slot
        float s3 = __shfl(hr3, col, 32);            // h3 -> upper-lane K3 slot
        v2f B1 = { lo ? hr0 : s2, lo ? hr1 : s3 };  // K rows: h0,h1,h2,h3
        v2f B2 = { lo ? hr4 : xA, xB };             // K rows: h4,x0,x1,x2

        v8f acc = __builtin_amdgcn_wmma_f32_16x16x4_f32(
            false, A2, false, B2, (short)0, cbias, false, false);
        acc = __builtin_amdgcn_wmma_f32_16x16x4_f32(
            false, A1, false, B1, (short)0, acc, false, false);

        hr0 = fast_tanh(acc[0]);
        hr1 = fast_tanh(acc[1]);
        hr2 = fast_tanh(acc[2]);
        hr3 = fast_tanh(acc[3]);
        hr4 = fast_tanh(acc[4]);

        if (lo) {   // 336 MB stream > 192 MB L2: write non-temporal
            float* op = out + ((size_t)t * BATCH + n) * HSZ;
            __builtin_nontemporal_store(hr0, op + 0);
            __builtin_nontemporal_store(hr1, op + 1);
            __builtin_nontemporal_store(hr2, op + 2);
            __builtin_nontemporal_store(hr3, op + 3);
            __builtin_nontemporal_store(hr4, op + 4);
        }
    }

    if (lo) {       // second tuple output: final hidden state
        float* fp = out + (size_t)S_LEN * BATCH * HSZ + (size_t)n * HSZ;
        fp[0] = hr0; fp[1] = hr1; fp[2] = hr2; fp[3] = hr3; fp[4] = hr4;
    }
}

extern "C" void kernel_launch(void* const* d_in, const int* in_sizes, int n_in,
                              void* d_out, int out_size, void* d_ws, size_t ws_size,
                              hipStream_t stream) {
    (void)in_sizes; (void)n_in; (void)out_size; (void)d_ws; (void)ws_size;
    const float* x   = (const float*)d_in[0];
    const float* hx0 = (const float*)d_in[1];
    const float* Wih = (const float*)d_in[2];
    const float* Whh = (const float*)d_in[3];
    const float* bih = (const float*)d_in[4];
    const float* bhh = (const float*)d_in[5];
    float* out = (float*)d_out;

    dim3 grid(BATCH / 16);   // 2048 single-wave workgroups, one 16-row tile each
    dim3 block(32);
    hipLaunchKernelGGL(rnn_scan_wmma, grid, block, 0, stream,
                       x, hx0, Wih, Whh, bih, bhh, out);
}